// AttentionMechanism_2121713844259
// MI455X (gfx1250) — compile-verified
//
#include <hip/hip_runtime.h>
#include <hip/hip_bf16.h>
#include <math.h>

// Problem constants (match reference)
#define C_SIZE 2048
#define BATCH  32
#define DIM    1024   // CONTEXT_DIM
#define HID    1024   // HIDDEN_SIZE
#define SEQ    64

#define NBLK   8                              // blocks per batch b
#define NWAVE  8                              // waves per block
#define KTOT   (NBLK * NWAVE)                 // 64 partial slots per b
#define CPW    (C_SIZE / KTOT)                // 32 context rows per wave

typedef float v2f __attribute__((ext_vector_type(2)));
typedef float v8f __attribute__((ext_vector_type(8)));

// ---------------------------------------------------------------------------
// Kernel 1: hidden_proj[b,d] = sum_h hidden[b,h] * W[d,h]   (h @ W^T)
// One wave per 16x16 output tile, fp32 WMMA 16x16x4, K-loop over HID.
// ---------------------------------------------------------------------------
__global__ __launch_bounds__(32) void proj_wmma_kernel(
    const float* __restrict__ h,   // [BATCH, HID]
    const float* __restrict__ W,   // [DIM, HID]
    float* __restrict__ hp)        // [BATCH, DIM]
{
    const int m0   = (blockIdx.x >> 6) * 16;   // 2 M-tiles (BATCH=32)
    const int n0   = (blockIdx.x & 63) * 16;   // 64 N-tiles (DIM=1024)
    const int lane = threadIdx.x;
    const int half = lane >> 4;                // 0 or 1 (K pair select)
    const int l16  = lane & 15;

    const float* arow = h + (size_t)(m0 + l16) * HID;  // A row m, over k
    const float* brow = W + (size_t)(n0 + l16) * HID;  // B col n = W row n, over k

    v8f acc = {};
#pragma unroll 4
    for (int k = 0; k < HID; k += 4) {
        const int kk = k + 2 * half;
        v2f a = *(const v2f*)(arow + kk);
        v2f b = *(const v2f*)(brow + kk);
        acc = __builtin_amdgcn_wmma_f32_16x16x4_f32(
            false, a, false, b, (short)0, acc, false, false);
    }

    float* out = hp + n0 + l16;
    const int mbase = m0 + 8 * half;
#pragma unroll
    for (int r = 0; r < 8; ++r)
        out[(size_t)(mbase + r) * DIM] = acc[r];
}

// ---------------------------------------------------------------------------
// Kernel 2 (fused, single pass over contextvects):
// For each (b, wave-slice of c): stream cv[c,b,:] once; compute score
// (wave32 shfl_xor dot reduce), record raw score, and do online-softmax
// accumulation of the weighted context vector entirely in registers.
// Emits per-wave partials (m, l, acc[1024]).
// ---------------------------------------------------------------------------
__global__ __launch_bounds__(256) void fused_scores_ctx_kernel(
    const float* __restrict__ cv,           // [C, B, D]
    const float* __restrict__ hp,           // [B, D]
    const unsigned char* __restrict__ mask, // [C, B] (bool)
    float* __restrict__ sc,                 // [C, B] raw scores
    float* __restrict__ pm,                 // [B, KTOT]
    float* __restrict__ pl,                 // [B, KTOT]
    float* __restrict__ pacc)               // [B, KTOT, D]
{
    const int b    = blockIdx.x >> 3;                         // 0..31
    const int k    = (blockIdx.x & 7) * NWAVE + (threadIdx.x >> 5); // 0..63
    const int lane = threadIdx.x & 31;

    // preload hp[b,:] slice: lane covers d = 4*lane + 128*i, i = 0..7
    const float4* hq = (const float4*)(hp + (size_t)b * DIM) + lane;
    float4 h[8];
#pragma unroll
    for (int i = 0; i < 8; ++i) h[i] = hq[i * 32];

    float4 acc[8];
#pragma unroll
    for (int i = 0; i < 8; ++i) acc[i] = make_float4(0.f, 0.f, 0.f, 0.f);
    float m = -INFINITY, l = 0.f;

    const int c0 = k * CPW;
    for (int ci = 0; ci < CPW; ++ci) {
        const int c = c0 + ci;
        const float4* p = (const float4*)(cv + ((size_t)c * BATCH + b) * DIM) + lane;
        if (ci + 2 < CPW)
            __builtin_prefetch(cv + ((size_t)(c + 2) * BATCH + b) * DIM + 4 * lane, 0, 0);

        float4 x[8];
#pragma unroll
        for (int i = 0; i < 8; ++i) x[i] = p[i * 32];

        float s = 0.f;
#pragma unroll
        for (int i = 0; i < 8; ++i)
            s += x[i].x * h[i].x + x[i].y * h[i].y + x[i].z * h[i].z + x[i].w * h[i].w;
        for (int off = 16; off; off >>= 1)
            s += __shfl_xor(s, off, 32);

        if (lane == 0) sc[(size_t)c * BATCH + b] = s;        // raw score
        if (mask[(size_t)c * BATCH + b]) s = -INFINITY;      // masked -> no contribution

        if (s != -INFINITY) {                                // wave-uniform branch
            if (s > m) {
                const float scale = __expf(m - s);           // rescale old state
                l *= scale;
#pragma unroll
                for (int i = 0; i < 8; ++i) {
                    acc[i].x *= scale; acc[i].y *= scale;
                    acc[i].z *= scale; acc[i].w *= scale;
                }
                m = s;
            }
            const float w = __expf(s - m);
            l += w;
#pragma unroll
            for (int i = 0; i < 8; ++i) {
                acc[i].x += w * x[i].x; acc[i].y += w * x[i].y;
                acc[i].z += w * x[i].z; acc[i].w += w * x[i].w;
            }
        }
    }

    if (lane == 0) {
        pm[b * KTOT + k] = m;
        pl[b * KTOT + k] = l;
    }
    float4* pout = (float4*)(pacc + ((size_t)b * KTOT + k) * DIM) + lane;
#pragma unroll
    for (int i = 0; i < 8; ++i) pout[i * 32] = acc[i];
}

// ---------------------------------------------------------------------------
// Kernel 3: exact column softmax from the recorded raw scores; writes the
// broadcast attentions output [SEQ, B, C] directly.
// ---------------------------------------------------------------------------
__global__ __launch_bounds__(256) void softmax_kernel(
    const float* __restrict__ sc,           // [C, B]
    const unsigned char* __restrict__ mask, // [C, B]
    float* __restrict__ out_attn)           // [SEQ, B, C]
{
    const int b    = blockIdx.x;
    const int tid  = threadIdx.x;
    const int wave = tid >> 5;
    const int lane = tid & 31;
    __shared__ float red[8];

    float v[8];
    float m = -INFINITY;
#pragma unroll
    for (int i = 0; i < 8; ++i) {
        const int c = tid + 256 * i;
        float s = sc[(size_t)c * BATCH + b];
        if (mask[(size_t)c * BATCH + b]) s = -INFINITY;
        v[i] = s;
        m = fmaxf(m, s);
    }
    for (int off = 16; off; off >>= 1)
        m = fmaxf(m, __shfl_xor(m, off, 32));
    if (lane == 0) red[wave] = m;
    __syncthreads();
    float bm = red[0];
#pragma unroll
    for (int i = 1; i < 8; ++i) bm = fmaxf(bm, red[i]);
    __syncthreads();

    float sum = 0.f;
#pragma unroll
    for (int i = 0; i < 8; ++i) {
        v[i] = __expf(v[i] - bm);
        sum += v[i];
    }
    for (int off = 16; off; off >>= 1)
        sum += __shfl_xor(sum, off, 32);
    if (lane == 0) red[wave] = sum;
    __syncthreads();
    float bs = red[0];
#pragma unroll
    for (int i = 1; i < 8; ++i) bs += red[i];
    const float inv = 1.f / bs;

#pragma unroll
    for (int i = 0; i < 8; ++i) {
        const int c = tid + 256 * i;
        const float w = v[i] * inv;
        for (int s = 0; s < SEQ; ++s)
            out_attn[(size_t)s * BATCH * C_SIZE + (size_t)b * C_SIZE + c] = w;
    }
}

// ---------------------------------------------------------------------------
// Kernel 4: log-sum-exp merge of the KTOT per-wave partials for each b,
// then broadcast context to [SEQ, B, D].
// ctx[b,d] = sum_k exp(m_k - M) * acc_k[d] / (sum_k l_k * exp(m_k - M))
// ---------------------------------------------------------------------------
__global__ __launch_bounds__(256) void merge_ctx_kernel(
    const float* __restrict__ pm,    // [B, KTOT]
    const float* __restrict__ pl,    // [B, KTOT]
    const float* __restrict__ pacc,  // [B, KTOT, D]
    float* __restrict__ out_ctx)     // [SEQ, B, D]
{
    const int b   = blockIdx.x;
    const int tid = threadIdx.x;
    __shared__ float sm[KTOT], sl[KTOT], sco[KTOT];

    if (tid < KTOT) {
        sm[tid] = pm[b * KTOT + tid];
        sl[tid] = pl[b * KTOT + tid];
    }
    __syncthreads();

    float M = -INFINITY;
#pragma unroll
    for (int k = 0; k < KTOT; ++k) M = fmaxf(M, sm[k]);
    if (tid < KTOT) sco[tid] = __expf(sm[tid] - M);
    __syncthreads();

    float L = 0.f;
#pragma unroll
    for (int k = 0; k < KTOT; ++k) L += sl[k] * sco[k];
    const float inv = 1.f / L;

    // each thread owns 4 d's: d = 4*tid
    const float4* pa = (const float4*)pacc + (size_t)b * KTOT * (DIM / 4) + tid;
    float4 s4 = make_float4(0.f, 0.f, 0.f, 0.f);
    for (int k = 0; k < KTOT; ++k) {
        const float4 a = pa[(size_t)k * (DIM / 4)];
        const float cf = sco[k];
        s4.x += cf * a.x; s4.y += cf * a.y; s4.z += cf * a.z; s4.w += cf * a.w;
    }
    s4.x *= inv; s4.y *= inv; s4.z *= inv; s4.w *= inv;

    for (int sq = 0; sq < SEQ; ++sq)
        ((float4*)out_ctx)[(size_t)sq * BATCH * (DIM / 4) + (size_t)b * (DIM / 4) + tid] = s4;
}

// ---------------------------------------------------------------------------
// Launch
// ---------------------------------------------------------------------------
extern "C" void kernel_launch(void* const* d_in, const int* in_sizes, int n_in,
                              void* d_out, int out_size, void* d_ws, size_t ws_size,
                              hipStream_t stream) {
    // inputs: [0]=seqlen(int,1) [1]=hidden f32 [2]=contextvects f32 [3]=W f32 [4]=padding_mask bool
    const float* hidden = (const float*)d_in[1];               // [1, B, H] == [B, H]
    const float* cv     = (const float*)d_in[2];               // [C, B, D]
    const float* W      = (const float*)d_in[3];               // [D, H]
    const unsigned char* mask = (const unsigned char*)d_in[4]; // [C, B]

    float* out = (float*)d_out;
    float* out_ctx  = out;                                     // [SEQ, B, D]
    float* out_attn = out + (size_t)SEQ * BATCH * DIM;         // [SEQ, B, C]

    float* ws = (float*)d_ws;
    float* hp   = ws;                                          // 32768 f
    float* sc   = hp + (size_t)BATCH * DIM;                    // 65536 f
    float* pm   = sc + (size_t)C_SIZE * BATCH;                 // 2048 f
    float* pl   = pm + (size_t)BATCH * KTOT;                   // 2048 f
    float* pacc = pl + (size_t)BATCH * KTOT;                   // 2097152 f (8 MB)

    proj_wmma_kernel<<<128, 32, 0, stream>>>(hidden, W, hp);
    fused_scores_ctx_kernel<<<BATCH * NBLK, 256, 0, stream>>>(cv, hp, mask, sc, pm, pl, pacc);
    softmax_kernel<<<BATCH, 256, 0, stream>>>(sc, mask, out_attn);
    merge_ctx_kernel<<<BATCH, 256, 0, stream>>>(pm, pl, pacc, out_ctx);
}